// DFineMultiscaleDeformableAttention_8839042695544
// MI455X (gfx1250) — compile-verified
//
#include <hip/hip_runtime.h>
#include <math.h>

typedef float v2f __attribute__((ext_vector_type(2)));
typedef float v8f __attribute__((ext_vector_type(8)));

#define BQ     4800   // B*Q = 16*300
#define HID    256
#define NHEADS 8
#define DHEAD  32
#define PTOT   16
#define NOFF   256    // cols of W_off  (= 128 points * 2)
#define NATT   128    // cols of W_attn (= 8 heads * 16 points)
#define STOT   13294  // sum of level sizes

// ---------------------------------------------------------------------------
// Kernel 1: fused GEMM  [4800 x 256] @ ([256 x 256] | [256 x 128]) + bias
// One wave32 computes one 16x16 output tile via V_WMMA_F32_16X16X4_F32.
// N tiles 0..15 -> W_off, 16..23 -> W_attn.  7200 tiles = 900 blocks * 8 waves.
// ---------------------------------------------------------------------------
__global__ __launch_bounds__(256) void dfine_gemm_kernel(
    const float* __restrict__ hs,   const float* __restrict__ Woff,
    const float* __restrict__ boff, const float* __restrict__ Watt,
    const float* __restrict__ batt, float* __restrict__ ws_off,
    float* __restrict__ ws_att)
{
    const int lane = threadIdx.x & 31;
    const int wave = threadIdx.x >> 5;
    const int tile = blockIdx.x * 8 + wave;   // 0..7199, wave-uniform
    const int tm   = tile / 24;               // M tile (0..299)
    const int tn   = tile % 24;               // N tile (0..23)
    const int half = lane >> 4;               // 0: lanes 0-15, 1: lanes 16-31
    const int l15  = lane & 15;

    const bool  isOff = (tn < 16);
    const float* W    = isOff ? Woff : Watt;
    const int   ncols = isOff ? NOFF : NATT;
    const int   col   = (isOff ? tn * 16 : (tn - 16) * 16) + l15;

    // A 16x4 f32 layout: lanes 0-15 -> K=k+0,k+1 ; lanes 16-31 -> K=k+2,k+3
    const float* arow = hs + (size_t)(tm * 16 + l15) * HID + 2 * half;
    // B 4x16 f32 layout (mirrored): row k+2*half and k+2*half+1 at this col
    const float* bcol = W + (size_t)(2 * half) * ncols + col;

    v8f c = {};
    for (int k = 0; k < HID; k += 4) {
        v2f a = *(const v2f*)(arow + k);
        v2f b;
        b.x = bcol[(size_t)k * ncols];
        b.y = bcol[(size_t)(k + 1) * ncols];
        c = __builtin_amdgcn_wmma_f32_16x16x4_f32(
                /*neg_a=*/false, a, /*neg_b=*/false, b,
                /*c_mod=*/(short)0, c, /*reuse_a=*/false, /*reuse_b=*/false);
    }

    const float bias = isOff ? boff[col] : batt[col];
    float* dst   = isOff ? (ws_off + col) : (ws_att + col);
    const int dn = isOff ? NOFF : NATT;
    // C/D layout: VGPR r -> M = r + 8*half, N = lane&15
#pragma unroll
    for (int r = 0; r < 8; ++r) {
        const int row = tm * 16 + r + 8 * half;
        dst[(size_t)row * dn] = c[r] + bias;
    }
}

// ---------------------------------------------------------------------------
// Kernel 2: softmax + bilinear sampling + weighted reduce.
// One wave32 per (b,q,h); lane = channel (d=32).  38400 waves = 4800 blocks.
// ---------------------------------------------------------------------------
__global__ __launch_bounds__(256) void dfine_sample_kernel(
    const float* __restrict__ enc,    const float* __restrict__ ref,
    const float* __restrict__ ws_off, const float* __restrict__ ws_att,
    float* __restrict__ out, float* __restrict__ aw_out)
{
    const int lane = threadIdx.x & 31;
    const int wave = threadIdx.x >> 5;
    const int id   = blockIdx.x * 8 + wave;  // (b*Q+q)*8 + h
    const int h    = id & 7;
    const int m    = id >> 3;                // b*Q + q  (0..4799)
    const int b    = m / 300;

    // ---- softmax over 16 logits (wave-uniform, recompute per lane) ----
    const float* lg = ws_att + (size_t)m * NATT + h * PTOT;
    float mx = lg[0];
#pragma unroll
    for (int p = 1; p < PTOT; ++p) mx = fmaxf(mx, lg[p]);
    float sum = 0.f;
#pragma unroll
    for (int p = 0; p < PTOT; ++p) sum += expf(lg[p] - mx);
    const float inv = 1.0f / sum;

    const float rx = ref[m * 4 + 0], ry = ref[m * 4 + 1];
    const float rw = ref[m * 4 + 2], rh = ref[m * 4 + 3];
    const float* off = ws_off + (size_t)m * NOFF + h * PTOT * 2;

    const int LW[4] = {100, 50, 25, 13};
    const int LS[4] = {0, 10000, 12500, 13125};

    const float* base = enc + (size_t)b * STOT * HID + h * DHEAD + lane;

    float acc = 0.f;
#pragma unroll
    for (int p = 0; p < PTOT; ++p) {
        const int lev = p >> 2;
        const int ww  = LW[lev];
        const int hh  = LW[lev];
        const float ap = expf(lg[p] - mx) * inv;

        const float lx = rx + off[2 * p + 0] * 0.25f * rw * 0.5f;
        const float ly = ry + off[2 * p + 1] * 0.25f * rh * 0.5f;
        const float x  = lx * (float)ww - 0.5f;
        const float y  = ly * (float)hh - 0.5f;
        const float x0 = floorf(x), y0 = floorf(y);
        const float wx1 = x - x0, wx0 = 1.f - wx1;
        const float wy1 = y - y0, wy0 = 1.f - wy1;

#pragma unroll
        for (int t = 0; t < 4; ++t) {
            const int dy = t >> 1, dx = t & 1;
            const float xx = x0 + (float)dx;
            const float yy = y0 + (float)dy;
            if (xx >= 0.f && xx <= (float)(ww - 1) &&
                yy >= 0.f && yy <= (float)(hh - 1)) {
                const int xi = (int)xx, yi = (int)yy;
                const float wt = (dy ? wy1 : wy0) * (dx ? wx1 : wx0);
                acc += ap * wt * base[(size_t)(LS[lev] + yi * ww + xi) * HID];
            }
        }
    }

    out[(size_t)m * HID + h * DHEAD + lane] = acc;
    if (lane < PTOT)
        aw_out[(size_t)m * NATT + h * PTOT + lane] = expf(lg[lane] - mx) * inv;
}

// ---------------------------------------------------------------------------
extern "C" void kernel_launch(void* const* d_in, const int* in_sizes, int n_in,
                              void* d_out, int out_size, void* d_ws, size_t ws_size,
                              hipStream_t stream) {
    const float* hs   = (const float*)d_in[0];  // hidden_states (16,300,256)
    const float* enc  = (const float*)d_in[1];  // encoder_hidden_states (16,13294,256)
    const float* ref  = (const float*)d_in[2];  // reference_points (16,300,1,4)
    const float* Woff = (const float*)d_in[3];  // (256,256)
    const float* boff = (const float*)d_in[4];  // (256,)
    const float* Watt = (const float*)d_in[5];  // (256,128)
    const float* batt = (const float*)d_in[6];  // (128,)

    float* out = (float*)d_out;                       // (16,300,256)
    float* aw  = out + (size_t)BQ * HID;              // (16,300,8,16)

    float* ws_off = (float*)d_ws;                     // (4800,256) raw offsets
    float* ws_att = ws_off + (size_t)BQ * NOFF;       // (4800,128) raw logits

    // 7200 tiles / 8 waves-per-block = 900 blocks exactly
    dfine_gemm_kernel<<<900, 256, 0, stream>>>(hs, Woff, boff, Watt, batt,
                                               ws_off, ws_att);
    // 38400 (b,q,h) waves / 8 = 4800 blocks exactly
    dfine_sample_kernel<<<4800, 256, 0, stream>>>(enc, ref, ws_off, ws_att,
                                                  out, aw);
}